// GCN_24953759989863
// MI455X (gfx1250) — compile-verified
//
#include <hip/hip_runtime.h>
#include <hip/hip_bf16.h>

typedef float v2f __attribute__((ext_vector_type(2)));
typedef float v8f __attribute__((ext_vector_type(8)));

#define N_NODES 50000
#define FEAT    64
#define M_TILES (N_NODES / 16)   // 3125 (exact)
#define N_TILES (FEAT / 16)      // 4
#define TOT_TILES (M_TILES * N_TILES)  // 12500

// ---------------------------------------------------------------------------
// Kernel 1: h = node @ W + b via V_WMMA_F32_16X16X4_F32.
// One wave per 16x16 output tile; K looped 64/4 = 16 WMMAs per tile.
// ---------------------------------------------------------------------------
__global__ void gcn_gemm_wmma(const float* __restrict__ node,
                              const float* __restrict__ W,
                              const float* __restrict__ bias,
                              float* __restrict__ h) {
    const int gwave = (int)((blockIdx.x * blockDim.x + threadIdx.x) >> 5);
    const int lane  = (int)(threadIdx.x & 31);
    if (gwave >= TOT_TILES) return;            // wave-uniform: EXEC stays all-1s at WMMA

    const int tileM = gwave / N_TILES;
    const int tileN = gwave - tileM * N_TILES;

    const int mrow = tileM * 16 + (lane & 15); // A: lane 0-15 and 16-31 both map M=0..15
    const int ncol = tileN * 16 + (lane & 15); // B/C/D: lanes hold N=0..15 per half
    const int kb   = (lane >> 4) * 2;          // upper half-wave carries K+2,K+3

    const float* __restrict__ arow = node + (size_t)mrow * FEAT;

    v8f c = {};
    #pragma unroll
    for (int k0 = 0; k0 < FEAT; k0 += 4) {
        v2f a, b;
        // A 16x4 tile: VGPR0 = K=kb, VGPR1 = K=kb+1 (contiguous in the row)
        const float2 av = *(const float2*)(arow + k0 + kb);
        a.x = av.x; a.y = av.y;
        // B 4x16 tile: VGPR0 = row K=kb of W, VGPR1 = row K=kb+1
        b.x = W[(size_t)(k0 + kb)     * FEAT + ncol];
        b.y = W[(size_t)(k0 + kb + 1) * FEAT + ncol];
        c = __builtin_amdgcn_wmma_f32_16x16x4_f32(
                /*neg_a=*/false, a, /*neg_b=*/false, b,
                /*c_mod=*/(short)0, c, /*reuse_a=*/false, /*reuse_b=*/false);
    }

    const float bv = bias[ncol];
    const int m0 = tileM * 16 + ((lane >> 4) << 3);  // rows r / r+8 split across half-waves
    #pragma unroll
    for (int r = 0; r < 8; ++r)
        h[(size_t)(m0 + r) * FEAT + ncol] = c[r] + bv;
}

// ---------------------------------------------------------------------------
// Kernel 2: per-edge gather * weight, scatter-add via L2 f32 atomics.
// One wave per edge, each lane owns 2 consecutive units (coalesced 2x128B).
// ---------------------------------------------------------------------------
__global__ void gcn_scatter(const float* __restrict__ h,
                            const float* __restrict__ ew,
                            const long long* __restrict__ eidx,
                            float* __restrict__ agg,
                            int nedges) {
    const int e    = (int)((blockIdx.x * blockDim.x + threadIdx.x) >> 5);
    const int lane = (int)(threadIdx.x & 31);
    if (e >= nedges) return;

    const long long s = eidx[(size_t)nedges + e];  // senders   = edge_index[1]
    const long long r = eidx[e];                   // receivers = edge_index[0]
    const float     w = ew[e];

    const float2 v = *(const float2*)(h + (size_t)s * FEAT + lane * 2);
    float* __restrict__ ap = agg + (size_t)r * FEAT + lane * 2;
    atomicAdd(ap,     v.x * w);   // global_atomic_add_f32, non-returning
    atomicAdd(ap + 1, v.y * w);
}

// ---------------------------------------------------------------------------
// Kernel 0/3: zero-init and in-place leaky_relu(0.2), float4-vectorized.
// ---------------------------------------------------------------------------
__global__ void gcn_zero4(float4* __restrict__ p, int n4) {
    const int i = (int)(blockIdx.x * blockDim.x + threadIdx.x);
    if (i < n4) p[i] = make_float4(0.f, 0.f, 0.f, 0.f);
}

__global__ void gcn_leaky4(float4* __restrict__ p, int n4) {
    const int i = (int)(blockIdx.x * blockDim.x + threadIdx.x);
    if (i < n4) {
        float4 v = p[i];
        v.x = v.x > 0.f ? v.x : 0.2f * v.x;
        v.y = v.y > 0.f ? v.y : 0.2f * v.y;
        v.z = v.z > 0.f ? v.z : 0.2f * v.z;
        v.w = v.w > 0.f ? v.w : 0.2f * v.w;
        p[i] = v;
    }
}

extern "C" void kernel_launch(void* const* d_in, const int* in_sizes, int n_in,
                              void* d_out, int out_size, void* d_ws, size_t ws_size,
                              hipStream_t stream) {
    const float*     node  = (const float*)d_in[0];
    const float*     edges = (const float*)d_in[1];
    const long long* eidx  = (const long long*)d_in[2];
    const float*     W     = (const float*)d_in[3];
    const float*     bias  = (const float*)d_in[4];
    float* out = (float*)d_out;
    float* h   = (float*)d_ws;                 // 50000*64*4 = 12.8 MB scratch

    const int nedges = in_sizes[1];            // edges is (E,1)
    const int ntot   = N_NODES * FEAT;
    const int n4     = ntot / 4;

    // 0) zero the accumulator (d_out is poisoned each timing round)
    gcn_zero4<<<(n4 + 255) / 256, 256, 0, stream>>>((float4*)out, n4);

    // 1) dense projection via f32 WMMA: 12500 tile-waves, 8 waves/block
    const int gemm_blocks = (TOT_TILES + 7) / 8;           // 1563
    gcn_gemm_wmma<<<gemm_blocks, 256, 0, stream>>>(node, W, bias, h);

    // 2) edge scatter: one wave per edge
    const long long scat_threads = (long long)nedges * 32;
    gcn_scatter<<<(unsigned)((scat_threads + 255) / 256), 256, 0, stream>>>(
        h, edges, eidx, out, nedges);

    // 3) leaky_relu(0.2) in place
    gcn_leaky4<<<(n4 + 255) / 256, 256, 0, stream>>>((float4*)out, n4);
}